// SamePadConv_16887811408386
// MI455X (gfx1250) — compile-verified
//
#include <hip/hip_runtime.h>
#include <hip/hip_bf16.h>

// ---------------- problem constants ----------------
#define Bb 8
#define CIN 64
#define COUT 64
#define CSKIP 256
#define Nn 325
#define Ll 256
#define LOUT 255
#define NH 64
#define OUT_ELEMS (Bb*COUT*Nn*LOUT)          // 42,432,000
#define SKIP_ELEMS ((long)Bb*CSKIP*Nn*LOUT)  // 169,728,000

typedef __attribute__((ext_vector_type(2))) float v2f;
typedef __attribute__((ext_vector_type(8))) float v8f;

__device__ __forceinline__ v8f wmma_f32(v2f a, v2f b, v8f c) {
  // D(16x16,f32) = A(16x4,f32) * B(4x16,f32) + C
  return __builtin_amdgcn_wmma_f32_16x16x4_f32(false, a, false, b, (short)0, c,
                                               false, false);
}

// ================= hypernetwork (tiny) =================
__global__ __launch_bounds__(256) void hyper_kernel(
    const float* __restrict__ grads, const float* __restrict__ conv_weight,
    const float* __restrict__ bias,
    const float* __restrict__ ctrl_w, const float* __restrict__ ctrl_b,
    const float* __restrict__ cw_w, const float* __restrict__ cw_b,
    const float* __restrict__ cb_w, const float* __restrict__ cb_b,
    const float* __restrict__ cf_w, const float* __restrict__ cf_b,
    const float* __restrict__ mlp_b,
    const float* __restrict__ bn_gamma, const float* __restrict__ bn_beta,
    const float* __restrict__ bn_mean, const float* __restrict__ bn_var,
    float* __restrict__ cw_out, float* __restrict__ bmod,
    float* __restrict__ fmod, float* __restrict__ bnscale,
    float* __restrict__ bnshift)
{
  __shared__ float rep[64 * 64];
  __shared__ float wmodS[128];
  const int t = threadIdx.x;
  for (int e = t; e < 64 * 64; e += 256) {
    int i = e >> 6, j = e & 63;
    const float* gi = grads + i * 128;
    const float* wj = ctrl_w + j * 128;
    float s = ctrl_b[j];
    for (int d = 0; d < 128; ++d) s = fmaf(gi[d], wj[d], s);
    rep[e] = s / (1.0f + expf(-s));  // silu
  }
  __syncthreads();
  if (t < 128) {  // w head: [CIN=64][K=2]
    int ci = t >> 1, k = t & 1;
    float s = cw_b[k];
    for (int j = 0; j < 64; ++j) s = fmaf(rep[ci * 64 + j], cw_w[k * 64 + j], s);
    wmodS[t] = s;
  } else if (t < 192) {  // b,f heads + BN fold, per channel
    int co = t - 128;
    float sb = cb_b[0], sf = cf_b[0];
    for (int j = 0; j < 64; ++j) {
      float r = rep[co * 64 + j];
      sb = fmaf(r, cb_w[j], sb);
      sf = fmaf(r, cf_w[j], sf);
    }
    bmod[co] = bias[co] * sb;
    fmod[co] = sf;
    float inv = rsqrtf(bn_var[co] + 1e-5f) * bn_gamma[co];
    bnscale[co] = inv;
    bnshift[co] = (mlp_b[co] - bn_mean[co]) * inv + bn_beta[co];
  }
  __syncthreads();
  for (int e = t; e < COUT * CIN * 2; e += 256)
    cw_out[e] = conv_weight[e] * wmodS[e & 127];
}

// ================= modulated temporal conv (WMMA, pipelined) =================
// xo[b,co,n,l] = fmod[co]*( sum_{kk<128} cw[co][kk]*x[b,kk>>1,n,l+(kk&1)] + bmod[co] )
__global__ __launch_bounds__(256) void conv_kernel(
    const float* __restrict__ x, const float* __restrict__ cw,
    const float* __restrict__ bmod, const float* __restrict__ fmod,
    float* __restrict__ xo)
{
  const int lane = threadIdx.x & 31;
  const int wave = blockIdx.x * 8 + (threadIdx.x >> 5);
  const int MT = COUT / 16, ST = 4;
  const int total = MT * ST * Bb * Nn;  // 41,600
  if (wave >= total) return;
  const int mt = wave % MT;
  const int st = (wave / MT) % ST;
  const int bn = wave / (MT * ST);
  const int n = bn % Nn;
  const int b = bn / Nn;
  const int co0 = mt * 16, l0 = st * 64;
  const int half = lane >> 4, lm = lane & 15;

  const float* xrow = x + ((b * CIN) * Nn + n) * (long)Ll + half * (Nn * Ll);
  int lcur[4], lnext[4];
#pragma unroll
  for (int t = 0; t < 4; ++t) {
    const int l = l0 + t * 16 + lm;           // <= 255: always a safe read
    lcur[t] = l;
    lnext[t] = (l + 1 < Ll) ? (l + 1) : l;    // clamped; pollutes masked col only
  }
  const v8f vzero = {0.f, 0.f, 0.f, 0.f, 0.f, 0.f, 0.f, 0.f};
  v8f acc[4] = {vzero, vzero, vzero, vzero};

  const float* cwp = cw + (co0 + lm) * 128 + 2 * half;
  // ---- software pipeline: prefetch next k-step while computing current ----
  v2f afc;
  float bxc[4], byc[4];
  afc.x = cwp[0];
  afc.y = cwp[1];
#pragma unroll
  for (int t = 0; t < 4; ++t) { bxc[t] = xrow[lcur[t]]; byc[t] = xrow[lnext[t]]; }
#pragma clang loop unroll_count(2)
  for (int k0 = 0; k0 < 124; k0 += 4) {
    const int kn = k0 + 4;
    v2f afn;
    float bxn[4], byn[4];
    afn.x = cwp[kn];
    afn.y = cwp[kn + 1];
    const float* xp = xrow + (kn >> 1) * (Nn * Ll);
#pragma unroll
    for (int t = 0; t < 4; ++t) { bxn[t] = xp[lcur[t]]; byn[t] = xp[lnext[t]]; }
#pragma unroll
    for (int t = 0; t < 4; ++t) {
      v2f bf; bf.x = bxc[t]; bf.y = byc[t];
      acc[t] = wmma_f32(afc, bf, acc[t]);
    }
    afc = afn;
#pragma unroll
    for (int t = 0; t < 4; ++t) { bxc[t] = bxn[t]; byc[t] = byn[t]; }
  }
#pragma unroll
  for (int t = 0; t < 4; ++t) {  // drain: k0 = 124
    v2f bf; bf.x = bxc[t]; bf.y = byc[t];
    acc[t] = wmma_f32(afc, bf, acc[t]);
  }
  float fm[8], bm[8];
#pragma unroll
  for (int r = 0; r < 8; ++r) {
    const int co = co0 + r + 8 * half;
    fm[r] = fmod[co];
    bm[r] = bmod[co];
  }
#pragma unroll
  for (int t = 0; t < 4; ++t) {
    const int ll = l0 + t * 16 + lm;
    if (ll < LOUT) {
      const float* ap = (const float*)&acc[t];
#pragma unroll
      for (int r = 0; r < 8; ++r) {
        const int co = co0 + r + 8 * half;
        xo[((b * COUT + co) * Nn + n) * (long)LOUT + ll] = fm[r] * (ap[r] + bm[r]);
      }
    }
  }
}

// ======== graph diffusion (WMMA, pipelined): Y = A^T X per (b,c) slice =======
// Y[m,l] = sum_n A[n,m] * X[n,l];  A: [325,325], X,Y: [325,255]
__global__ __launch_bounds__(256) void diffuse_kernel(
    const float* __restrict__ A, const float* __restrict__ X,
    float* __restrict__ Y)
{
  const int lane = threadIdx.x & 31;
  const int wave = blockIdx.x * 8 + (threadIdx.x >> 5);
  const int MT = 21, ST = 4;
  const int total = Bb * COUT * MT * ST;  // 43,008
  if (wave >= total) return;
  const int st = wave % ST;
  const int mt = (wave / ST) % MT;
  const int slice = wave / (MT * ST);
  const int m0 = mt * 16, l0 = st * 64;
  const int half = lane >> 4, lm = lane & 15;

  const float* Xs = X + slice * (long)(Nn * LOUT);
  float* Ys = Y + slice * (long)(Nn * LOUT);
  const int mm = m0 + lm;
  const int mmc = (mm < Nn) ? mm : (Nn - 1);  // clamp: pollutes masked rows only
  int llc[4];
#pragma unroll
  for (int t = 0; t < 4; ++t) {
    const int l = l0 + t * 16 + lm;
    llc[t] = (l < LOUT) ? l : (LOUT - 1);     // clamp: pollutes masked cols only
  }
  const v8f vzero = {0.f, 0.f, 0.f, 0.f, 0.f, 0.f, 0.f, 0.f};
  v8f acc[4] = {vzero, vzero, vzero, vzero};

  const float* Ap = A + (2 * half) * Nn + mmc;   // row (k0 + 2*half), col mmc
  const float* Bp0 = Xs + (2 * half) * LOUT;
  // ---- software pipeline over 81 unconditional k-steps ----
  v2f afc;
  float bxc[4], byc[4];
  afc.x = Ap[0];
  afc.y = Ap[Nn];
#pragma unroll
  for (int t = 0; t < 4; ++t) { bxc[t] = Bp0[llc[t]]; byc[t] = Bp0[LOUT + llc[t]]; }
#pragma clang loop unroll_count(2)
  for (int k0 = 0; k0 < 320; k0 += 4) {  // compute k0, prefetch k0+4 (<= 320)
    const int kn = k0 + 4;
    v2f afn;
    float bxn[4], byn[4];
    afn.x = Ap[kn * Nn];
    afn.y = Ap[kn * Nn + Nn];
    const float* Bp = Bp0 + kn * LOUT;
#pragma unroll
    for (int t = 0; t < 4; ++t) { bxn[t] = Bp[llc[t]]; byn[t] = Bp[LOUT + llc[t]]; }
#pragma unroll
    for (int t = 0; t < 4; ++t) {
      v2f bf; bf.x = bxc[t]; bf.y = byc[t];
      acc[t] = wmma_f32(afc, bf, acc[t]);
    }
    afc = afn;
#pragma unroll
    for (int t = 0; t < 4; ++t) { bxc[t] = bxn[t]; byc[t] = byn[t]; }
  }
#pragma unroll
  for (int t = 0; t < 4; ++t) {  // drain: k0 = 320 (rows 320..323)
    v2f bf; bf.x = bxc[t]; bf.y = byc[t];
    acc[t] = wmma_f32(afc, bf, acc[t]);
  }
  {  // tail: k = 324 (valid only in half==0 lanes); zero A's k-slot kills products
    const float km = (half == 0) ? 1.f : 0.f;
    const int krow = (half == 0) ? 324 : 0;      // safe address either way
    v2f af;
    af.x = km * A[krow * Nn + mmc];
    af.y = 0.f;
    const float* Bp = Xs + krow * LOUT;
#pragma unroll
    for (int t = 0; t < 4; ++t) {
      v2f bf;
      bf.x = Bp[llc[t]];
      bf.y = 0.f;
      acc[t] = wmma_f32(af, bf, acc[t]);
    }
  }
  const bool mfull = (m0 + 15 < Nn);
#pragma unroll
  for (int t = 0; t < 4; ++t) {
    const int ll = l0 + t * 16 + lm;
    if (ll < LOUT) {
      const float* ap = (const float*)&acc[t];
      if (mfull) {
#pragma unroll
        for (int r = 0; r < 8; ++r)
          Ys[(m0 + r + 8 * half) * (long)LOUT + ll] = ap[r];
      } else {
#pragma unroll
        for (int r = 0; r < 8; ++r) {
          const int m = m0 + r + 8 * half;
          if (m < Nn) Ys[m * (long)LOUT + ll] = ap[r];
        }
      }
    }
  }
}

// ========== 1x1 channel-mix GEMM (WMMA, pipelined): skip & mlp groups ========
// dst[b,o,n,l] (+)= sum_{c<64} W[o][c0w+c] * src[b,c,n,l]  (+ bias[o]) (+ addsrc)
__global__ __launch_bounds__(256) void chanmix_kernel(
    const float* __restrict__ src, const float* __restrict__ W, int wpitch,
    int c0w, const float* __restrict__ biasv, const float* __restrict__ addsrc,
    int Laux, int aoff, int O, int accflag, float* __restrict__ dst)
{
  const int lane = threadIdx.x & 31;
  const int wave = blockIdx.x * 8 + (threadIdx.x >> 5);
  const int MT = O / 16, ST = 4;
  const int total = MT * ST * Bb * Nn;
  if (wave >= total) return;
  const int mt = wave % MT;
  const int st = (wave / MT) % ST;
  const int bn = wave / (MT * ST);
  const int n = bn % Nn;
  const int b = bn / Nn;
  const int o0 = mt * 16, l0 = st * 64;
  const int half = lane >> 4, lm = lane & 15;

  const int NL = Nn * LOUT;
  const float* srow = src + ((b * 64) * Nn + n) * (long)LOUT;
  float* drow = dst + ((long)(b * O) * Nn + n) * LOUT;
  int llc[4];
#pragma unroll
  for (int t = 0; t < 4; ++t) {
    const int l = l0 + t * 16 + lm;
    llc[t] = (l < LOUT) ? l : (LOUT - 1);  // clamp: pollutes masked cols only
  }
  const v8f vzero = {0.f, 0.f, 0.f, 0.f, 0.f, 0.f, 0.f, 0.f};
  v8f acc[4] = {vzero, vzero, vzero, vzero};
  if (accflag) {
#pragma unroll
    for (int t = 0; t < 4; ++t) {
      float* ap = (float*)&acc[t];
#pragma unroll
      for (int r = 0; r < 8; ++r)
        ap[r] = drow[(o0 + r + 8 * half) * (long)NL + llc[t]];
    }
  }
  const float* Wp = W + (o0 + lm) * wpitch + c0w + 2 * half;
  const float* s0 = srow + (2 * half) * NL;
  // ---- software pipeline over 16 k-steps ----
  v2f afc;
  float bxc[4], byc[4];
  afc.x = Wp[0];
  afc.y = Wp[1];
#pragma unroll
  for (int t = 0; t < 4; ++t) { bxc[t] = s0[llc[t]]; byc[t] = s0[NL + llc[t]]; }
#pragma clang loop unroll_count(2)
  for (int c0 = 0; c0 < 60; c0 += 4) {
    const int cn = c0 + 4;
    v2f afn;
    float bxn[4], byn[4];
    afn.x = Wp[cn];
    afn.y = Wp[cn + 1];
    const float* sp = s0 + cn * NL;
#pragma unroll
    for (int t = 0; t < 4; ++t) { bxn[t] = sp[llc[t]]; byn[t] = sp[NL + llc[t]]; }
#pragma unroll
    for (int t = 0; t < 4; ++t) {
      v2f bf; bf.x = bxc[t]; bf.y = byc[t];
      acc[t] = wmma_f32(afc, bf, acc[t]);
    }
    afc = afn;
#pragma unroll
    for (int t = 0; t < 4; ++t) { bxc[t] = bxn[t]; byc[t] = byn[t]; }
  }
#pragma unroll
  for (int t = 0; t < 4; ++t) {  // drain: c0 = 60
    v2f bf; bf.x = bxc[t]; bf.y = byc[t];
    acc[t] = wmma_f32(afc, bf, acc[t]);
  }
#pragma unroll
  for (int t = 0; t < 4; ++t) {
    const int ll = l0 + t * 16 + lm;
    if (ll < LOUT) {
      const float* ap = (const float*)&acc[t];
#pragma unroll
      for (int r = 0; r < 8; ++r) {
        const int o = o0 + r + 8 * half;
        float v = ap[r];
        if (biasv) v += biasv[o];
        if (addsrc) v += addsrc[((b * O + o) * Nn + n) * (long)Laux + aoff + ll];
        drow[o * (long)NL + ll] = v;
      }
    }
  }
}

// ================= epilogue: residual + BatchNorm (folded) ===================
__global__ __launch_bounds__(256) void bn_kernel(
    const float* __restrict__ x, const float* __restrict__ bnscale,
    const float* __restrict__ bnshift, float* __restrict__ out)
{
  const long idx = (long)blockIdx.x * 256 + threadIdx.x;
  if (idx >= (long)OUT_ELEMS) return;
  const int l = (int)(idx % LOUT);
  long r = idx / LOUT;
  const int n = (int)(r % Nn);
  r /= Nn;
  const int c = (int)(r % COUT);
  const int b = (int)(r / COUT);
  const float xv = x[((b * CIN + c) * Nn + n) * (long)Ll + l + 1];
  out[idx] = (out[idx] + xv) * bnscale[c] + bnshift[c];
}

// ============================ launch =========================================
extern "C" void kernel_launch(void* const* d_in, const int* in_sizes, int n_in,
                              void* d_out, int out_size, void* d_ws,
                              size_t ws_size, hipStream_t stream) {
  const float* x           = (const float*)d_in[0];
  const float* skip        = (const float*)d_in[1];
  const float* sup0        = (const float*)d_in[2];
  const float* sup1        = (const float*)d_in[3];
  const float* grads       = (const float*)d_in[4];
  const float* conv_weight = (const float*)d_in[5];
  const float* bias        = (const float*)d_in[6];
  const float* ctrl_w      = (const float*)d_in[7];
  const float* ctrl_b      = (const float*)d_in[8];
  const float* cw_w        = (const float*)d_in[9];
  const float* cw_b        = (const float*)d_in[10];
  const float* cb_w        = (const float*)d_in[11];
  const float* cb_b        = (const float*)d_in[12];
  const float* cf_w        = (const float*)d_in[13];
  const float* cf_b        = (const float*)d_in[14];
  const float* skip_w      = (const float*)d_in[15];
  const float* skip_b      = (const float*)d_in[16];
  const float* mlp_w       = (const float*)d_in[17];
  const float* mlp_b       = (const float*)d_in[18];
  const float* bn_gamma    = (const float*)d_in[19];
  const float* bn_beta     = (const float*)d_in[20];
  const float* bn_mean     = (const float*)d_in[21];
  const float* bn_var      = (const float*)d_in[22];

  float* ws      = (float*)d_ws;
  float* cw      = ws;            // 8192
  float* bmod    = ws + 8192;     // 64
  float* fmod    = ws + 8256;     // 64
  float* bnscale = ws + 8320;     // 64
  float* bnshift = ws + 8384;     // 64
  float* xo      = ws + 8448;     // 42,432,000

  float* outp    = (float*)d_out;              // [42,432,000] final "out"
  float* skipout = outp + (long)OUT_ELEMS;     // [169,728,000] final "skip_out"
  // reuse the skip_out region as scratch for diffusion temporaries;
  // the skip GEMM (which only reads xo) runs LAST and overwrites it.
  float* t1 = skipout;
  float* t2 = skipout + (long)OUT_ELEMS;

  hyper_kernel<<<1, 256, 0, stream>>>(grads, conv_weight, bias, ctrl_w, ctrl_b,
                                      cw_w, cw_b, cb_w, cb_b, cf_w, cf_b,
                                      mlp_b, bn_gamma, bn_beta, bn_mean, bn_var,
                                      cw, bmod, fmod, bnscale, bnshift);

  {  // modulated conv -> xo : 4*4*2600 = 41,600 waves
    const int waves = 4 * 4 * Bb * Nn;
    conv_kernel<<<(waves + 7) / 8, 256, 0, stream>>>(x, cw, bmod, fmod, xo);
  }

  const int dwaves = Bb * COUT * 21 * 4;  // 43,008
  const int dblk = (dwaves + 7) / 8;
  const int mwaves64 = 4 * 4 * Bb * Nn;   // 41,600
  const int mblk64 = (mwaves64 + 7) / 8;

  // mlp accumulation, 5 groups of 64 channels, directly into "out" region
  chanmix_kernel<<<mblk64, 256, 0, stream>>>(xo, mlp_w, 320, 0, nullptr,
                                             nullptr, 0, 0, 64, 0, outp);
  diffuse_kernel<<<dblk, 256, 0, stream>>>(sup0, xo, t1);
  chanmix_kernel<<<mblk64, 256, 0, stream>>>(t1, mlp_w, 320, 64, nullptr,
                                             nullptr, 0, 0, 64, 1, outp);
  diffuse_kernel<<<dblk, 256, 0, stream>>>(sup0, t1, t2);
  chanmix_kernel<<<mblk64, 256, 0, stream>>>(t2, mlp_w, 320, 128, nullptr,
                                             nullptr, 0, 0, 64, 1, outp);
  diffuse_kernel<<<dblk, 256, 0, stream>>>(sup1, xo, t1);
  chanmix_kernel<<<mblk64, 256, 0, stream>>>(t1, mlp_w, 320, 192, nullptr,
                                             nullptr, 0, 0, 64, 1, outp);
  diffuse_kernel<<<dblk, 256, 0, stream>>>(sup1, t1, t2);
  chanmix_kernel<<<mblk64, 256, 0, stream>>>(t2, mlp_w, 320, 256, nullptr,
                                             nullptr, 0, 0, 64, 1, outp);

  // residual + BN epilogue (in-place on out region)
  bn_kernel<<<(OUT_ELEMS + 255) / 256, 256, 0, stream>>>(x, bnscale, bnshift,
                                                         outp);

  // skip branch last: overwrites t1/t2 scratch with real skip_out
  {
    const int waves = 16 * 4 * Bb * Nn;  // O=256 -> 166,400 waves
    chanmix_kernel<<<(waves + 7) / 8, 256, 0, stream>>>(
        xo, skip_w, 64, 0, skip_b, skip, Ll, 1, 256, 0, skipout);
  }
}